// GATEncoder_69810398429750
// MI455X (gfx1250) — compile-verified
//
#include <hip/hip_runtime.h>
#include <cstdint>

typedef __attribute__((ext_vector_type(2))) float v2f;
typedef __attribute__((ext_vector_type(8))) float v8f;
typedef __attribute__((ext_vector_type(4))) unsigned int v4u;
typedef __attribute__((ext_vector_type(4))) int v4i;
typedef __attribute__((ext_vector_type(8))) int v8i;

#define NNODES 50000
#define NEDGES 800000
#define FDIM   128
#define LDS_PITCH 129          // 128 + 1 DWORD TDM pad -> bank-conflict-free B reads
#define RT_PER_BLOCK 5         // 3125 row tiles = 625 blocks * 5

static inline int cdiv(long long a, long long b) { return (int)((a + b - 1) / b); }

// ---------- monotone float<->uint mapping for atomic float max ----------
__device__ __forceinline__ unsigned flip_f32(float f) {
  unsigned u = __float_as_uint(f);
  return (u & 0x80000000u) ? ~u : (u | 0x80000000u);
}
__device__ __forceinline__ float unflip_f32(unsigned u) {
  return (u & 0x80000000u) ? __uint_as_float(u & 0x7fffffffu) : __uint_as_float(~u);
}
#define FLIP_NEG_INF 0x007FFFFFu   // flip_f32(-inf)

// ---------- init kernels ----------
__global__ void k_fill_u32(unsigned* __restrict__ p, long long n, unsigned v) {
  long long i = (long long)blockIdx.x * blockDim.x + threadIdx.x;
  if (i < n) p[i] = v;
}
__global__ void k_fill_f32(float* __restrict__ p, long long n, float v) {
  long long i = (long long)blockIdx.x * blockDim.x + threadIdx.x;
  if (i < n) p[i] = v;
}
__global__ void k_bias_bcast(float* __restrict__ p, const float* __restrict__ bias,
                             long long n, int F) {
  long long i = (long long)blockIdx.x * blockDim.x + threadIdx.x;
  if (i < n) p[i] = bias[(int)(i % F)];
}

// ---------- H = A @ W^T via TDM-staged W + V_WMMA_F32_16X16X4_F32 ----------
// W (128x128 f32, 64KB) is DMA'd into LDS once per workgroup by the Tensor Data
// Mover with a 1-DWORD pad per 128-DWORD row (pitch 129 floats). 8 waves each own
// one 16-wide column tile and sweep RT_PER_BLOCK row tiles, so one TDM transfer
// feeds 40 WMMA tiles. A streams from global; B comes from LDS.
__global__ void k_gemm_xwt_wmma(const float* __restrict__ A, const float* __restrict__ W,
                                float* __restrict__ Out, int M) {
  __shared__ float shW[FDIM * LDS_PITCH];

  if (threadIdx.x == 0) {
    unsigned lds_off = (unsigned)(unsigned long long)(uintptr_t)(&shW[0]);
    unsigned long long ga = (unsigned long long)(uintptr_t)W;
    // ---- D# group 0: count=1, lds_addr, global_addr[56:0], type=2 ----
    v4u g0;
    g0[0] = 1u;
    g0[1] = lds_off;
    g0[2] = (unsigned)(ga & 0xFFFFFFFFull);
    g0[3] = (unsigned)((ga >> 32) & 0x01FFFFFFull) | (2u << 30);
    // ---- D# group 1: data_size=4B, pad 1 DWORD per 128 DWORDs,
    //      tensor 128x128, tile 128x128, dim0 stride 128 ----
    v8i g1;
    g1[0] = (int)((2u << 16) | (1u << 20) | (6u << 22));  // data_size | pad_en | pad_interval=128
    g1[1] = (int)(128u << 16);   // tensor_dim0 = 128   (bits 63:48)
    g1[2] = (int)(128u << 16);   // tensor_dim1 = 128   (bits 95:80)
    g1[3] = (int)(128u << 16);   // tile_dim0  = 128    (bits 127:112)
    g1[4] = 128;                 // tile_dim1  = 128    (bits 143:128)
    g1[5] = 128;                 // tensor_dim0_stride  (bits 191:160)
    g1[6] = 0;
    g1[7] = 0;
    v4i gz = {0, 0, 0, 0};
#if __clang_major__ >= 23
    v8i gz8 = {0, 0, 0, 0, 0, 0, 0, 0};
    __builtin_amdgcn_tensor_load_to_lds(g0, g1, gz, gz, gz8, 0);
#else
    __builtin_amdgcn_tensor_load_to_lds(g0, g1, gz, gz, 0);
#endif
  }
  __builtin_amdgcn_s_wait_tensorcnt((short)0);
  __syncthreads();

  const int lane = threadIdx.x & 31;
  const int wave = threadIdx.x >> 5;              // column tile 0..7
  const int rowB = (wave << 4) + (lane & 15);     // N index (row of W) this lane feeds
  const int kof  = (lane >> 4) << 1;              // 0 or 2

  for (int rt = 0; rt < RT_PER_BLOCK; ++rt) {
    const int tm = blockIdx.x * RT_PER_BLOCK + rt;
    if (tm * 16 >= M) return;                     // wave-uniform: EXEC stays all-ones
    const int rowA = tm * 16 + (lane & 15);
    const float* __restrict__ aRow = A + (size_t)rowA * FDIM;

    v8f acc = {0.f, 0.f, 0.f, 0.f, 0.f, 0.f, 0.f, 0.f};
#pragma unroll 4
    for (int k0 = 0; k0 < FDIM; k0 += 4) {
      v2f a = *reinterpret_cast<const v2f*>(aRow + k0 + kof);
      v2f b;
      b.x = shW[rowB * LDS_PITCH + k0 + kof];
      b.y = shW[rowB * LDS_PITCH + k0 + kof + 1];
      acc = __builtin_amdgcn_wmma_f32_16x16x4_f32(false, a, false, b,
                                                  (short)0, acc, false, false);
    }
    const int outRow0 = tm * 16 + ((lane >> 4) << 3);
    const int outCol  = (wave << 4) + (lane & 15);
#pragma unroll
    for (int j = 0; j < 8; ++j)
      Out[(size_t)(outRow0 + j) * FDIM + outCol] = acc[j];
  }
}

// ---------- per-node attention logits: as[n,h]=<h[n,h,:],a_src[h,:]> ----------
__global__ void k_alpha(const float* __restrict__ h, const float* __restrict__ a_src,
                        const float* __restrict__ a_dst, float* __restrict__ as_,
                        float* __restrict__ ad_, int Nn, int H, int C) {
  int gid = blockIdx.x * blockDim.x + threadIdx.x;
  if (gid >= Nn * H) return;
  int n = gid / H, hh = gid % H;
  const float* hp = h + (size_t)n * H * C + (size_t)hh * C;
  const float* s = a_src + (size_t)hh * C;
  const float* d = a_dst + (size_t)hh * C;
  float s1 = 0.f, s2 = 0.f;
  for (int c = 0; c < C; ++c) { float v = hp[c]; s1 = fmaf(v, s[c], s1); s2 = fmaf(v, d[c], s2); }
  as_[gid] = s1;
  ad_[gid] = s2;
}

// ---------- edge pass 1: segment max of leaky_relu(as[src]+ad[dst]) ----------
__global__ void k_edge_max(const long long* __restrict__ ei, const float* __restrict__ as_,
                           const float* __restrict__ ad_, unsigned* __restrict__ mflip,
                           int E, int Nn, int H, int lgH) {
  long long gid = (long long)blockIdx.x * blockDim.x + threadIdx.x;
  long long total = (long long)(E + Nn) << lgH;
  if (gid >= total) return;
  int e = (int)(gid >> lgH), hh = (int)(gid & (H - 1));
  int src = (e < E) ? (int)ei[e] : (e - E);
  int dst = (e < E) ? (int)ei[(size_t)E + e] : (e - E);
  float v = as_[(size_t)src * H + hh] + ad_[(size_t)dst * H + hh];
  v = (v > 0.f) ? v : 0.2f * v;  // leaky_relu, slope 0.2
  atomicMax(&mflip[(size_t)dst * H + hh], flip_f32(v));
}

// ---------- edge pass 2: p = exp(e - m[dst]); s[dst] += p ----------
__global__ void k_edge_exp(const long long* __restrict__ ei, const float* __restrict__ as_,
                           const float* __restrict__ ad_, const unsigned* __restrict__ mflip,
                           float* __restrict__ pbuf, float* __restrict__ sbuf,
                           int E, int Nn, int H, int lgH) {
  long long gid = (long long)blockIdx.x * blockDim.x + threadIdx.x;
  long long total = (long long)(E + Nn) << lgH;
  if (gid >= total) return;
  int e = (int)(gid >> lgH), hh = (int)(gid & (H - 1));
  int src = (e < E) ? (int)ei[e] : (e - E);
  int dst = (e < E) ? (int)ei[(size_t)E + e] : (e - E);
  float v = as_[(size_t)src * H + hh] + ad_[(size_t)dst * H + hh];
  v = (v > 0.f) ? v : 0.2f * v;
  float m = unflip_f32(mflip[(size_t)dst * H + hh]);
  float p = __expf(v - m);
  pbuf[gid] = p;
  atomicAdd(&sbuf[(size_t)dst * H + hh], p);
}

// ---------- edge pass 3: wave-per-edge, float4 per lane ----------
// out[dst,f] += h[src,f] * p/(s[dst]+eps); 32 lanes x 4 floats = 128 features.
__global__ void k_edge_aggr(const long long* __restrict__ ei, const float* __restrict__ h,
                            const float* __restrict__ pbuf, const float* __restrict__ sbuf,
                            float* __restrict__ out, int E, int Nn, int H, int C) {
  long long gid = (long long)blockIdx.x * blockDim.x + threadIdx.x;
  long long eIdx = gid >> 5;
  if (eIdx >= (long long)E + Nn) return;
  int e = (int)eIdx;
  int lane = threadIdx.x & 31;
  int src = (e < E) ? (int)ei[e] : (e - E);
  int dst = (e < E) ? (int)ei[(size_t)E + e] : (e - E);
  int f0 = lane << 2;              // 4 consecutive features (within one head: C % 4 == 0)
  int hh = f0 / C;
  float a = pbuf[(size_t)e * H + hh] / (sbuf[(size_t)dst * H + hh] + 1e-16f);
  const float4 hv = *reinterpret_cast<const float4*>(h + (size_t)src * FDIM + f0);
  float* op = out + (size_t)dst * FDIM + f0;
  atomicAdd(op + 0, hv.x * a);
  atomicAdd(op + 1, hv.y * a);
  atomicAdd(op + 2, hv.z * a);
  atomicAdd(op + 3, hv.w * a);
}

// ---------- elementwise ELU (alpha=1) ----------
__global__ void k_elu(float* __restrict__ p, long long n) {
  long long i = (long long)blockIdx.x * blockDim.x + threadIdx.x;
  if (i < n) { float v = p[i]; p[i] = (v > 0.f) ? v : (__expf(v) - 1.f); }
}

extern "C" void kernel_launch(void* const* d_in, const int* in_sizes, int n_in,
                              void* d_out, int out_size, void* d_ws, size_t ws_size,
                              hipStream_t stream) {
  (void)in_sizes; (void)n_in; (void)out_size; (void)ws_size;
  const float*     x    = (const float*)d_in[0];
  const long long* ei   = (const long long*)d_in[1];
  const float*     W1   = (const float*)d_in[2];
  const float*     aS1  = (const float*)d_in[3];
  const float*     aD1  = (const float*)d_in[4];
  const float*     b1   = (const float*)d_in[5];
  const float*     W2   = (const float*)d_in[6];
  const float*     aS2  = (const float*)d_in[7];
  const float*     aD2  = (const float*)d_in[8];
  const float*     b2   = (const float*)d_in[9];
  float* out = (float*)d_out;

  const int N = NNODES, E = NEDGES, F = FDIM;
  const long long Etot = (long long)E + N;

  // workspace layout (floats)
  float* wsf  = (float*)d_ws;
  float* hbuf = wsf;                              // N*F   transformed features
  float* agg  = hbuf + (size_t)N * F;             // N*F   aggregation / layer-2 input
  float* as_  = agg  + (size_t)N * F;             // N*4
  float* ad_  = as_  + (size_t)N * 4;             // N*4
  float* mbuf = ad_  + (size_t)N * 4;             // N*4   (uint-flipped max)
  float* sbuf = mbuf + (size_t)N * 4;             // N*4
  float* pbuf = sbuf + (size_t)N * 4;             // Etot*4
  unsigned* mflip = (unsigned*)mbuf;

  const int BT = 256;
  const int gemmBlocks = cdiv(N / 16, RT_PER_BLOCK);   // 625

  // ================= Layer 1: GATConv(128 -> 32, heads=4, concat) =================
  {
    const int H = 4, C = 32, lgH = 2;
    k_gemm_xwt_wmma<<<gemmBlocks, BT, 0, stream>>>(x, W1, hbuf, N);
    k_alpha<<<cdiv((long long)N * H, BT), BT, 0, stream>>>(hbuf, aS1, aD1, as_, ad_, N, H, C);
    k_fill_u32<<<cdiv((long long)N * H, BT), BT, 0, stream>>>(mflip, (long long)N * H, FLIP_NEG_INF);
    k_fill_f32<<<cdiv((long long)N * H, BT), BT, 0, stream>>>(sbuf, (long long)N * H, 0.f);
    k_bias_bcast<<<cdiv((long long)N * F, BT), BT, 0, stream>>>(agg, b1, (long long)N * F, F);
    k_edge_max <<<cdiv(Etot * H, BT), BT, 0, stream>>>(ei, as_, ad_, mflip, E, N, H, lgH);
    k_edge_exp <<<cdiv(Etot * H, BT), BT, 0, stream>>>(ei, as_, ad_, mflip, pbuf, sbuf, E, N, H, lgH);
    k_edge_aggr<<<cdiv(Etot * 32, BT), BT, 0, stream>>>(ei, hbuf, pbuf, sbuf, agg, E, N, H, C);
    k_elu<<<cdiv((long long)N * F, BT), BT, 0, stream>>>(agg, (long long)N * F);
  }

  // ================= Layer 2: GATConv(128 -> 128, heads=1, mean) =================
  {
    const int H = 1, C = 128, lgH = 0;
    k_gemm_xwt_wmma<<<gemmBlocks, BT, 0, stream>>>(agg, W2, hbuf, N);
    k_alpha<<<cdiv((long long)N * H, BT), BT, 0, stream>>>(hbuf, aS2, aD2, as_, ad_, N, H, C);
    k_fill_u32<<<cdiv((long long)N * H, BT), BT, 0, stream>>>(mflip, (long long)N * H, FLIP_NEG_INF);
    k_fill_f32<<<cdiv((long long)N * H, BT), BT, 0, stream>>>(sbuf, (long long)N * H, 0.f);
    k_bias_bcast<<<cdiv((long long)N * F, BT), BT, 0, stream>>>(out, b2, (long long)N * F, F);
    k_edge_max <<<cdiv(Etot * H, BT), BT, 0, stream>>>(ei, as_, ad_, mflip, E, N, H, lgH);
    k_edge_exp <<<cdiv(Etot * H, BT), BT, 0, stream>>>(ei, as_, ad_, mflip, pbuf, sbuf, E, N, H, lgH);
    k_edge_aggr<<<cdiv(Etot * 32, BT), BT, 0, stream>>>(ei, hbuf, pbuf, sbuf, out, E, N, H, C);
    // heads=1 mean == identity; bias2 pre-loaded into out before atomics.
  }
}